// TensorProductFFN_50208167690495
// MI455X (gfx1250) — compile-verified
//
#include <hip/hip_runtime.h>

#define NNODES 50000
#define NEDGES 800000
#define SDIM 128
#define HDIM 256
#define ECDIM 128
#define GINDIM 512
#define DTOT 9
#define NB 64
#define MT 4  // 16-row M tiles per block
#define LMAXV 2

typedef float v2f __attribute__((ext_vector_type(2)));
typedef float v8f __attribute__((ext_vector_type(8)));

#define WMMA4(A, B, C) \
  __builtin_amdgcn_wmma_f32_16x16x4_f32(false, (A), false, (B), (short)0, (C), false, false)

__device__ __forceinline__ float silu_f(float x) {
  return x * __builtin_amdgcn_rcpf(1.0f + __expf(-x));
}

// ---------------- kernel 1: zero scatter workspace ----------------
__global__ void k_zero(float* __restrict__ p, int n) {
  int i = blockIdx.x * blockDim.x + threadIdx.x;
  if (i < n) p[i] = 0.0f;
}

// ---------------- kernel 2: edge scatter-add ----------------
__global__ void k_scatter(const float* __restrict__ t_ij,
                          const float* __restrict__ a_ij,
                          const int* __restrict__ eidx,
                          float* __restrict__ ci, float* __restrict__ ca,
                          float* __restrict__ cnt) {
  int i = blockIdx.x * blockDim.x + threadIdx.x;
  int e = i >> 5;
  if (e >= NEDGES) return;
  int c4 = (i & 31) << 2;
  int dst = eidx[NEDGES + e];  // edge_index row 1
  size_t src = (size_t)e * ECDIM + c4;
  float4 tv = *(const float4*)(t_ij + src);
  float4 av = *(const float4*)(a_ij + src);
  float* cip = ci + (size_t)dst * ECDIM + c4;
  float* cap = ca + (size_t)dst * ECDIM + c4;
  atomicAdd(cip + 0, tv.x); atomicAdd(cip + 1, tv.y);
  atomicAdd(cip + 2, tv.z); atomicAdd(cip + 3, tv.w);
  atomicAdd(cap + 0, av.x); atomicAdd(cap + 1, av.y);
  atomicAdd(cap + 2, av.z); atomicAdd(cap + 3, av.w);
  if (c4 == 0) atomicAdd(cnt + dst, 1.0f);
}

// ---------------- kernel 3: sum -> mean ----------------
__global__ void k_mean(float* __restrict__ ci, float* __restrict__ ca,
                       const float* __restrict__ cnt) {
  int i = blockIdx.x * blockDim.x + threadIdx.x;
  if (i >= NNODES * ECDIM) return;
  int n = i >> 7;
  float inv = 1.0f / fmaxf(cnt[n], 1.0f);
  ci[i] *= inv;
  ca[i] *= inv;
}

// ---------------- kernel 4: fused gate+FFN, 64 nodes / block ----------------
// 8 wave32s, dynamic LDS (256 KB): gate_in 128K | tmp 64K | g 64K.
// Each wave owns 4 (M) x 2 (N) 16x16 tiles for H=256 GEMMs, 4 (M) x 1 for
// the final S=128 GEMM: 8 independent v_wmma_f32_16x16x4_f32 per k-step.
__global__ __launch_bounds__(256) void k_fused(
    const float* __restrict__ x,
    const float* __restrict__ ci, const float* __restrict__ ca,
    const float* __restrict__ W1, const float* __restrict__ W2,
    const float* __restrict__ Wg1, const float* __restrict__ bg1,
    const float* __restrict__ Wg2, const float* __restrict__ bg2,
    float* __restrict__ out) {
  extern __shared__ float smem[];
  float* s_gin = smem;                // [NB][GINDIM]
  float* s_tmp = smem + NB * GINDIM;  // [NB][HDIM]  (gate hidden, then h*g)
  float* s_g   = s_tmp + NB * HDIM;   // [NB][HDIM]

  const int tid = threadIdx.x;
  const int lane = tid & 31;
  const int wv = tid >> 5;
  const int r = lane & 15;
  const int kh = lane >> 4;
  const int n0 = blockIdx.x * NB;
  const int nvalid = min(NB, NNODES - n0);

  auto cnode = [&](int row) { return n0 + (row < nvalid ? row : nvalid - 1); };

  // fill l-invariant gate_in sections: c_i, c_ang, x0 (clamped rows for tail)
  for (int i = tid; i < NB * SDIM; i += 256) {
    int row = i >> 7, s = i & 127;
    size_t nb = (size_t)cnode(row);
    s_gin[row * GINDIM + s] = ci[nb * ECDIM + s];
    s_gin[row * GINDIM + ECDIM + s] = ca[nb * ECDIM + s];
    s_gin[row * GINDIM + 2 * ECDIM + SDIM + s] = x[(nb * DTOT + 0) * SDIM + s];
  }

  int off = 0;
  for (int l = 0; l <= LMAXV; ++l) {
    const int deg = 2 * l + 1;
    const float* Wg1l = Wg1 + (size_t)l * GINDIM * HDIM;
    const float* bg1l = bg1 + l * HDIM;
    const float* Wg2l = Wg2 + (size_t)l * HDIM * HDIM;
    const float* bg2l = bg2 + l * HDIM;

    // pull this l's gate panels toward the caches while norms compute
    for (int pb = tid; pb < (GINDIM * HDIM) / 64; pb += 256)
      __builtin_prefetch(Wg1l + pb * 64, 0, 1);
    for (int pb = tid; pb < (HDIM * HDIM) / 64; pb += 256)
      __builtin_prefetch(Wg2l + pb * 64, 0, 1);

    // norm_l
    for (int i = tid; i < NB * SDIM; i += 256) {
      int row = i >> 7, s = i & 127;
      float acc = 0.0f;
      size_t base = ((size_t)cnode(row) * DTOT + off) * SDIM + s;
      for (int dd = 0; dd < deg; ++dd) {
        float v = x[base + (size_t)dd * SDIM];
        acc += v * v;
      }
      s_gin[row * GINDIM + 2 * ECDIM + s] = sqrtf(acc);
    }
    __syncthreads();

    // ---- gate GEMM1: (64x512)@(512x256)+b -> silu -> s_tmp ----
    {
      const int c0 = wv * 32 + r, c1 = c0 + 16;
      float b0 = bg1l[c0], b1 = bg1l[c1];
      v8f accA[MT], accB[MT];
#pragma unroll
      for (int m = 0; m < MT; ++m)
        for (int v = 0; v < 8; ++v) { accA[m][v] = b0; accB[m][v] = b1; }
      for (int k0 = 0; k0 < GINDIM; k0 += 4) {
        int ka = k0 + 2 * kh;
        v2f af[MT], bb0, bb1;
#pragma unroll
        for (int m = 0; m < MT; ++m) {
          af[m].x = s_gin[(m * 16 + r) * GINDIM + ka];
          af[m].y = s_gin[(m * 16 + r) * GINDIM + ka + 1];
        }
        bb0.x = Wg1l[(size_t)ka * HDIM + c0]; bb0.y = Wg1l[(size_t)(ka + 1) * HDIM + c0];
        bb1.x = Wg1l[(size_t)ka * HDIM + c1]; bb1.y = Wg1l[(size_t)(ka + 1) * HDIM + c1];
#pragma unroll
        for (int m = 0; m < MT; ++m) {
          accA[m] = WMMA4(af[m], bb0, accA[m]);
          accB[m] = WMMA4(af[m], bb1, accB[m]);
        }
      }
#pragma unroll
      for (int m = 0; m < MT; ++m)
        for (int v = 0; v < 8; ++v) {
          int row = m * 16 + v + 8 * kh;
          s_tmp[row * HDIM + c0] = silu_f(accA[m][v]);
          s_tmp[row * HDIM + c1] = silu_f(accB[m][v]);
        }
    }
    __syncthreads();

    // ---- gate GEMM2: (64x256)@(256x256)+b -> silu -> s_g ----
    {
      const int c0 = wv * 32 + r, c1 = c0 + 16;
      float b0 = bg2l[c0], b1 = bg2l[c1];
      v8f accA[MT], accB[MT];
#pragma unroll
      for (int m = 0; m < MT; ++m)
        for (int v = 0; v < 8; ++v) { accA[m][v] = b0; accB[m][v] = b1; }
      for (int k0 = 0; k0 < HDIM; k0 += 4) {
        int ka = k0 + 2 * kh;
        v2f af[MT], bb0, bb1;
#pragma unroll
        for (int m = 0; m < MT; ++m) {
          af[m].x = s_tmp[(m * 16 + r) * HDIM + ka];
          af[m].y = s_tmp[(m * 16 + r) * HDIM + ka + 1];
        }
        bb0.x = Wg2l[(size_t)ka * HDIM + c0]; bb0.y = Wg2l[(size_t)(ka + 1) * HDIM + c0];
        bb1.x = Wg2l[(size_t)ka * HDIM + c1]; bb1.y = Wg2l[(size_t)(ka + 1) * HDIM + c1];
#pragma unroll
        for (int m = 0; m < MT; ++m) {
          accA[m] = WMMA4(af[m], bb0, accA[m]);
          accB[m] = WMMA4(af[m], bb1, accB[m]);
        }
      }
#pragma unroll
      for (int m = 0; m < MT; ++m)
        for (int v = 0; v < 8; ++v) {
          int row = m * 16 + v + 8 * kh;
          s_g[row * HDIM + c0] = silu_f(accA[m][v]);
          s_g[row * HDIM + c1] = silu_f(accB[m][v]);
        }
    }
    __syncthreads();

    // ---- per degree slot: h = x_d @ W1, act, *g, then @ W2 -> out ----
    for (int dd = 0; dd < deg; ++dd) {
      const int d = off + dd;
      {  // h GEMM (64x128)@(128x256), gated into s_tmp
        const int c0 = wv * 32 + r, c1 = c0 + 16;
        v8f accA[MT], accB[MT];
#pragma unroll
        for (int m = 0; m < MT; ++m) { accA[m] = {}; accB[m] = {}; }
        size_t arow[MT];
#pragma unroll
        for (int m = 0; m < MT; ++m)
          arow[m] = ((size_t)cnode(m * 16 + r) * DTOT + d) * SDIM;
        for (int k0 = 0; k0 < SDIM; k0 += 4) {
          int ka = k0 + 2 * kh;
          v2f af[MT], bb0, bb1;
#pragma unroll
          for (int m = 0; m < MT; ++m) {
            af[m].x = x[arow[m] + ka];
            af[m].y = x[arow[m] + ka + 1];
          }
          bb0.x = W1[(size_t)ka * HDIM + c0]; bb0.y = W1[(size_t)(ka + 1) * HDIM + c0];
          bb1.x = W1[(size_t)ka * HDIM + c1]; bb1.y = W1[(size_t)(ka + 1) * HDIM + c1];
#pragma unroll
          for (int m = 0; m < MT; ++m) {
            accA[m] = WMMA4(af[m], bb0, accA[m]);
            accB[m] = WMMA4(af[m], bb1, accB[m]);
          }
        }
#pragma unroll
        for (int m = 0; m < MT; ++m)
          for (int v = 0; v < 8; ++v) {
            int row = m * 16 + v + 8 * kh;
            float h0 = accA[m][v], h1 = accB[m][v];
            if (l == 0) { h0 = silu_f(h0); h1 = silu_f(h1); }
            s_tmp[row * HDIM + c0] = h0 * s_g[row * HDIM + c0];
            s_tmp[row * HDIM + c1] = h1 * s_g[row * HDIM + c1];
          }
      }
      __syncthreads();
      {  // out GEMM (64x256)@(256x128) -> global (guarded tail stores)
        const int c = wv * 16 + r;
        v8f acc[MT];
#pragma unroll
        for (int m = 0; m < MT; ++m) acc[m] = {};
        for (int k0 = 0; k0 < HDIM; k0 += 4) {
          int ka = k0 + 2 * kh;
          v2f af[MT], bb;
#pragma unroll
          for (int m = 0; m < MT; ++m) {
            af[m].x = s_tmp[(m * 16 + r) * HDIM + ka];
            af[m].y = s_tmp[(m * 16 + r) * HDIM + ka + 1];
          }
          bb.x = W2[(size_t)ka * SDIM + c]; bb.y = W2[(size_t)(ka + 1) * SDIM + c];
#pragma unroll
          for (int m = 0; m < MT; ++m) acc[m] = WMMA4(af[m], bb, acc[m]);
        }
#pragma unroll
        for (int m = 0; m < MT; ++m)
          for (int v = 0; v < 8; ++v) {
            int row = m * 16 + v + 8 * kh;
            if (row < nvalid)
              out[((size_t)(n0 + row) * DTOT + d) * SDIM + c] = acc[m][v];
          }
      }
      __syncthreads();
    }
    off += deg;
  }
}

extern "C" void kernel_launch(void* const* d_in, const int* in_sizes, int n_in,
                              void* d_out, int out_size, void* d_ws, size_t ws_size,
                              hipStream_t stream) {
  (void)in_sizes; (void)n_in; (void)out_size; (void)ws_size;
  const float* x_emb = (const float*)d_in[0];
  const float* t_ij  = (const float*)d_in[1];
  const float* a_ij  = (const float*)d_in[2];
  const int*   eidx  = (const int*)d_in[3];
  const float* W1    = (const float*)d_in[4];
  const float* W2    = (const float*)d_in[5];
  const float* Wg1   = (const float*)d_in[6];
  const float* bg1   = (const float*)d_in[7];
  const float* Wg2   = (const float*)d_in[8];
  const float* bg2   = (const float*)d_in[9];
  float* out = (float*)d_out;

  // workspace: ci (N*EC) | ca (N*EC) | cnt (N)
  float* ci  = (float*)d_ws;
  float* ca  = ci + (size_t)NNODES * ECDIM;
  float* cnt = ca + (size_t)NNODES * ECDIM;

  int nzero = 2 * NNODES * ECDIM + NNODES;
  k_zero<<<(nzero + 255) / 256, 256, 0, stream>>>(ci, nzero);

  int nscat = NEDGES * 32;
  k_scatter<<<(nscat + 255) / 256, 256, 0, stream>>>(t_ij, a_ij, eidx, ci, ca, cnt);

  int nmean = NNODES * ECDIM;
  k_mean<<<(nmean + 255) / 256, 256, 0, stream>>>(ci, ca, cnt);

  const int smem_bytes = (NB * GINDIM + 2 * NB * HDIM) * sizeof(float);  // 256 KB
  k_fused<<<(NNODES + NB - 1) / NB, 256, smem_bytes, stream>>>(
      x_emb, ci, ca, W1, W2, Wg1, bg1, Wg2, bg2, out);
}